// MMPA_65893388255265
// MI455X (gfx1250) — compile-verified
//
#include <hip/hip_runtime.h>
#include <hip/hip_bf16.h>
#include <math.h>
#include <stdint.h>

typedef float v2f __attribute__((ext_vector_type(2)));
typedef float v8f __attribute__((ext_vector_type(8)));

#define B_     16
#define CIN    256
#define CMID   32
#define HW     128
#define PIX    (HW*HW)     // 16384
#define NPIX   64          // pixels per block tile
#define COUT   256
#define KROWS  64          // k-rows per TDM phase (4 phases cover K=256)

// ---------------------------------------------------------------------------
// Tensor Data Mover support (gfx1250). 2D tile load: `rows` lines of `width`
// f32 elements, line stride `stride` elements, HBM -> LDS, row-major in LDS.
// D# per CDNA5 ISA. Guarded: manual-copy fallback keeps
// the WMMA path alive if the builtin is absent.
// ---------------------------------------------------------------------------
#if defined(__has_builtin)
#if __has_builtin(__builtin_amdgcn_tensor_load_to_lds) && \
    __has_builtin(__builtin_amdgcn_s_wait_tensorcnt)
#define HAVE_TDM 1
#endif
#endif
#ifndef HAVE_TDM
#define HAVE_TDM 0
#endif

#if HAVE_TDM
typedef unsigned int u32x4 __attribute__((ext_vector_type(4)));
typedef int          i32x4 __attribute__((ext_vector_type(4)));
typedef int          i32x8 __attribute__((ext_vector_type(8)));

__device__ __forceinline__ void tdm_load_2d(uint32_t lds_byte_addr, uint64_t gaddr,
                                            uint32_t width, uint32_t rows,
                                            uint64_t stride_elts)
{
  u32x4 g0;
  g0.x = 1u;                                                  // count=1 (valid), user mode
  g0.y = lds_byte_addr;                                       // D#.lds_addr
  g0.z = (uint32_t)gaddr;                                     // global_addr[31:0]
  g0.w = (uint32_t)((gaddr >> 32) & 0x01FFFFFFu) | (2u << 30);// addr[56:32] | type=2
  i32x8 g1;
  g1[0] = (int)(2u << 16);                                    // data_size=2 (4 bytes)
  g1[1] = (int)((width & 0xFFFFu) << 16);                     // tensor_dim0[15:0] @bits63:48
  g1[2] = (int)(((width >> 16) & 0xFFFFu) |                   // tensor_dim0[31:16]
                ((rows & 0xFFFFu) << 16));                    // tensor_dim1[15:0]
  g1[3] = (int)(((rows >> 16) & 0xFFFFu) |                    // tensor_dim1[31:16]
                ((width & 0xFFFFu) << 16));                   // tile_dim0
  g1[4] = (int)(rows & 0xFFFFu);                              // tile_dim1 (tile_dim2=0)
  g1[5] = (int)(uint32_t)(stride_elts & 0xFFFFFFFFu);         // tensor_dim0_stride lo
  g1[6] = (int)(uint32_t)((stride_elts >> 32) & 0xFFFFu);     // tensor_dim0_stride hi
  g1[7] = 0;                                                  // tensor_dim1_stride (2D)
  i32x4 gz = {0, 0, 0, 0};
#if __clang_major__ >= 23
  i32x8 gz8 = {0, 0, 0, 0, 0, 0, 0, 0};
  __builtin_amdgcn_tensor_load_to_lds(g0, g1, gz, gz, gz8, 0);
#else
  __builtin_amdgcn_tensor_load_to_lds(g0, g1, gz, gz, 0);
#endif
}
#endif // HAVE_TDM

// ---------------------------------------------------------------------------
// Kernel 1: proj 1x1 conv as GEMM  z[32, pix] = PW[32,256] @ X[256, pix] + pb
// 4-phase TDM pipeline, double-buffered 16KB B tiles (row-major [k][64]).
// 8 waves, each owns one 16x16 C-tile; 64 v_wmma_f32_16x16x4_f32 per wave.
// LDS: A 32KB (swizzled, one ds_load_b64 per fragment) + 2x16KB B = 64KB.
// ---------------------------------------------------------------------------
__device__ __forceinline__ void proj_gemm_phase(v8f& acc, const float* ldsA,
                                                const float* buf, int phase,
                                                int m0, int n0, int half, int l16)
{
  #pragma unroll 4
  for (int kg = 0; kg < KROWS / 4; ++kg) {
    v2f a = *(const v2f*)&ldsA[(phase * (KROWS/4) + kg) * (CMID * 4) + (m0 + l16) * 4 + half * 2];
    const int krow = kg * 4 + half * 2;
    v2f bf;
    bf.x = buf[(krow    ) * NPIX + n0 + l16];
    bf.y = buf[(krow + 1) * NPIX + n0 + l16];
    acc = __builtin_amdgcn_wmma_f32_16x16x4_f32(false, a, false, bf,
                                                (short)0, acc, false, false);
  }
}

__global__ __launch_bounds__(256) void proj_kernel(
    const float* __restrict__ x, const float* __restrict__ pw,
    const float* __restrict__ pb, float* __restrict__ z)
{
  __shared__ float ldsA[64 * CMID * 4];     // [kgrp 0..63][m 0..31][k%4]  32KB
  __shared__ float ldsB[2][KROWS * NPIX];   // two row-major [k][p] tiles 2x16KB

  const int blk = blockIdx.x;
  const int b   = blk >> 8;                 // 256 tiles per batch
  const int p0  = (blk & 255) * NPIX;
  const int t   = threadIdx.x;

  // Stage A once: pw[m][k] -> ldsA[(k>>2)*128 + m*4 + (k&3)]
  for (int i = t; i < CMID * CIN; i += 256) {
    int m = i >> 8, k = i & 255;
    ldsA[(k >> 2) * (CMID * 4) + m * 4 + (k & 3)] = pw[m * CIN + k];
  }

  const float* xb = x + ((size_t)b * CIN) * PIX + p0;

#if HAVE_TDM
  const uint32_t ldsB_base = (uint32_t)(uintptr_t)(void*)&ldsB[0][0];
#define STAGE_B(bufidx, rowoff) \
  do { if (t < 32) tdm_load_2d(ldsB_base + (uint32_t)(bufidx) * (KROWS * NPIX * 4), \
        (uint64_t)(uintptr_t)(xb + (size_t)(rowoff) * PIX), NPIX, KROWS, PIX); } while (0)
#define WAIT_B(n) do { if (t < 32) __builtin_amdgcn_s_wait_tensorcnt(n); } while (0)
#else
#define STAGE_B(bufidx, rowoff) \
  do { for (int i = t; i < KROWS * NPIX; i += 256) { int k = i >> 6, p = i & 63; \
        ldsB[bufidx][k * NPIX + p] = xb[((size_t)(rowoff) + k) * PIX + p]; } } while (0)
#define WAIT_B(n) do { } while (0)
#endif

  const int wave = t >> 5, lane = t & 31;
  const int half = lane >> 4, l16 = lane & 15;
  const int m0 = (wave & 1) * 16;           // which 16 of the 32 out channels
  const int n0 = (wave >> 1) * 16;          // which 16 of the 64 pixels

  // Prime the pipeline: two tiles in flight (TENSORcnt = 2).
  STAGE_B(0, 0 * KROWS);
  STAGE_B(1, 1 * KROWS);

  v8f acc = {};
  // phase 0
  WAIT_B(1); __syncthreads();
  proj_gemm_phase(acc, ldsA, ldsB[0], 0, m0, n0, half, l16);
  __syncthreads();
  STAGE_B(0, 2 * KROWS);
  // phase 1
  WAIT_B(1); __syncthreads();
  proj_gemm_phase(acc, ldsA, ldsB[1], 1, m0, n0, half, l16);
  __syncthreads();
  STAGE_B(1, 3 * KROWS);
  // phase 2
  WAIT_B(1); __syncthreads();
  proj_gemm_phase(acc, ldsA, ldsB[0], 2, m0, n0, half, l16);
  __syncthreads();
  // phase 3
  WAIT_B(0); __syncthreads();
  proj_gemm_phase(acc, ldsA, ldsB[1], 3, m0, n0, half, l16);

#undef STAGE_B
#undef WAIT_B

  // C/D layout: vgpr r, lane -> M = r + 8*(lane/16), N = lane%16
  float* zb = z + ((size_t)b * CMID) * PIX + p0;
  #pragma unroll
  for (int r = 0; r < 8; ++r) {
    int c = m0 + r + 8 * half;
    zb[(size_t)c * PIX + n0 + l16] = acc[r] + pb[c];
  }
}

// ---------------------------------------------------------------------------
// Kernel 2: spatial mean of z per (b, c). z is L2-resident (33.5MB < 192MB).
// ---------------------------------------------------------------------------
__global__ __launch_bounds__(256) void mean_kernel(const float* __restrict__ z,
                                                   float* __restrict__ g)
{
  const int bc = blockIdx.x;                  // bc = b*32 + c
  const float* zp = z + (size_t)bc * PIX;
  float s = 0.f;
  for (int i = threadIdx.x; i < PIX; i += 256) s += zp[i];
  #pragma unroll
  for (int off = 16; off > 0; off >>= 1) s += __shfl_down(s, off, 32);
  __shared__ float red[8];
  if ((threadIdx.x & 31) == 0) red[threadIdx.x >> 5] = s;
  __syncthreads();
  if (threadIdx.x == 0) {
    float v = 0.f;
    #pragma unroll
    for (int i = 0; i < 8; ++i) v += red[i];
    g[bc] = v * (1.f / (float)PIX);
  }
}

// ---------------------------------------------------------------------------
// Kernel 3: gating MLP: h = gelu(g @ fc1^T + b1); w = softmax(h @ fc2^T + b2)
// ---------------------------------------------------------------------------
__global__ void gate_kernel(const float* __restrict__ g,
                            const float* __restrict__ fc1w, const float* __restrict__ fc1b,
                            const float* __restrict__ fc2w, const float* __restrict__ fc2b,
                            float* __restrict__ wgate)
{
  int b = threadIdx.x;
  if (b >= B_) return;
  float h[16];
  #pragma unroll
  for (int j = 0; j < 16; ++j) {
    float acc = fc1b[j];
    for (int c = 0; c < CMID; ++c) acc += g[b * CMID + c] * fc1w[j * CMID + c];
    h[j] = 0.5f * acc * (1.f + erff(acc * 0.70710678118654752f));
  }
  float l[3]; float mx = -1e30f;
  #pragma unroll
  for (int k = 0; k < 3; ++k) {
    float acc = fc2b[k];
    for (int j = 0; j < 16; ++j) acc += h[j] * fc2w[k * 16 + j];
    l[k] = acc; mx = fmaxf(mx, acc);
  }
  float se = 0.f;
  #pragma unroll
  for (int k = 0; k < 3; ++k) { l[k] = __expf(l[k] - mx); se += l[k]; }
  float inv = 1.f / se;
  #pragma unroll
  for (int k = 0; k < 3; ++k) wgate[b * 3 + k] = l[k] * inv;
}

// ---------------------------------------------------------------------------
// Kernel 4: dilated depthwise (d=3,5,7, zero-pad) + gated combine + residual,
// then reproj 1x1 conv as GEMM out[256, pix] = RW[256,32] @ fused[32, pix] + rb.
// Depthwise taps read z directly from global (L2-resident); fused tile lands
// in an 8KB LDS B-operand; 8 waves each run 8 C-tiles (64 WMMAs).
// LDS: A 32KB + B 8KB = 40KB.
// ---------------------------------------------------------------------------
__global__ __launch_bounds__(256) void fuse_reproj_kernel(
    const float* __restrict__ z,
    const float* __restrict__ e1w, const float* __restrict__ e1b,
    const float* __restrict__ e2w, const float* __restrict__ e2b,
    const float* __restrict__ e3w, const float* __restrict__ e3b,
    const float* __restrict__ wg,
    const float* __restrict__ rw,  const float* __restrict__ rb,
    float* __restrict__ out)
{
  __shared__ float ldsA[8 * COUT * 4];   // [kgrp 0..7][m 0..255][k%4]  32KB
  __shared__ float ldsB[8 * NPIX * 4];   // [kgrp 0..7][p 0..63][k%4]    8KB

  const int blk = blockIdx.x;
  const int b   = blk >> 8;
  const int p0  = (blk & 255) * NPIX;
  const int y   = p0 >> 7;               // tile lives in a single row
  const int x0  = p0 & 127;
  const int t   = threadIdx.x;

  // Stage reproj weights: rw[m][k] -> ldsA[(k>>2)*1024 + m*4 + (k&3)]
  for (int i = t; i < COUT * CMID; i += 256) {
    int m = i >> 5, k = i & 31;
    ldsA[(k >> 2) * (COUT * 4) + m * 4 + (k & 3)] = rw[m * CMID + k];
  }

  const float w0 = wg[b * 3 + 0], w1 = wg[b * 3 + 1], w2 = wg[b * 3 + 2];

  // Each thread: channel c = t/8, 8 contiguous pixels starting at (t%8)*8.
  const int c = t >> 3;
  const int pbase = (t & 7) * 8;
  const float* zc = z + ((size_t)(b * CMID + c)) * PIX;
  float k1[9], k2[9], k3[9];
  #pragma unroll
  for (int i = 0; i < 9; ++i) {
    k1[i] = e1w[c * 9 + i]; k2[i] = e2w[c * 9 + i]; k3[i] = e3w[c * 9 + i];
  }
  const float bias = w0 * e1b[c] + w1 * e2b[c] + w2 * e3b[c];

  for (int j = 0; j < 8; ++j) {
    const int p  = pbase + j;
    const int gx = x0 + p;
    float a1 = 0.f, a2 = 0.f, a3 = 0.f;
    #pragma unroll
    for (int ky = 0; ky < 3; ++ky) {
      #pragma unroll
      for (int kx = 0; kx < 3; ++kx) {
        { int yy = y + (ky - 1) * 3, xx = gx + (kx - 1) * 3;
          if ((unsigned)yy < HW && (unsigned)xx < HW) a1 += k1[ky*3+kx] * zc[yy*HW + xx]; }
        { int yy = y + (ky - 1) * 5, xx = gx + (kx - 1) * 5;
          if ((unsigned)yy < HW && (unsigned)xx < HW) a2 += k2[ky*3+kx] * zc[yy*HW + xx]; }
        { int yy = y + (ky - 1) * 7, xx = gx + (kx - 1) * 7;
          if ((unsigned)yy < HW && (unsigned)xx < HW) a3 += k3[ky*3+kx] * zc[yy*HW + xx]; }
      }
    }
    float fused = w0 * a1 + w1 * a2 + w2 * a3 + bias + zc[y * HW + gx];
    ldsB[(c >> 2) * (NPIX * 4) + p * 4 + (c & 3)] = fused;
  }
  __syncthreads();

  const int wave = t >> 5, lane = t & 31;
  const int half = lane >> 4, l16 = lane & 15;
  float* ob = out + ((size_t)b * COUT) * PIX + p0;
  #pragma unroll
  for (int mt = 0; mt < 2; ++mt) {
    const int m0 = wave * 32 + mt * 16;       // 8 waves cover M=256
    #pragma unroll
    for (int nt = 0; nt < 4; ++nt) {
      const int n0 = nt * 16;
      v8f acc = {};
      #pragma unroll
      for (int kg = 0; kg < 8; ++kg) {        // K = 32
        v2f a  = *(const v2f*)&ldsA[kg * (COUT * 4) + (m0 + l16) * 4 + half * 2];
        v2f bf = *(const v2f*)&ldsB[kg * (NPIX * 4) + (n0 + l16) * 4 + half * 2];
        acc = __builtin_amdgcn_wmma_f32_16x16x4_f32(false, a, false, bf,
                                                    (short)0, acc, false, false);
      }
      #pragma unroll
      for (int r = 0; r < 8; ++r) {
        int cc = m0 + r + 8 * half;
        ob[(size_t)cc * PIX + n0 + l16] = acc[r] + rb[cc];
      }
    }
  }
}

// ---------------------------------------------------------------------------
extern "C" void kernel_launch(void* const* d_in, const int* in_sizes, int n_in,
                              void* d_out, int out_size, void* d_ws, size_t ws_size,
                              hipStream_t stream)
{
  const float* x    = (const float*)d_in[0];
  const float* pw   = (const float*)d_in[1];
  const float* pb   = (const float*)d_in[2];
  const float* e1w  = (const float*)d_in[3];
  const float* e1b  = (const float*)d_in[4];
  const float* e2w  = (const float*)d_in[5];
  const float* e2b  = (const float*)d_in[6];
  const float* e3w  = (const float*)d_in[7];
  const float* e3b  = (const float*)d_in[8];
  const float* fc1w = (const float*)d_in[9];
  const float* fc1b = (const float*)d_in[10];
  const float* fc2w = (const float*)d_in[11];
  const float* fc2b = (const float*)d_in[12];
  const float* rw   = (const float*)d_in[13];
  const float* rb   = (const float*)d_in[14];
  float* out = (float*)d_out;

  // Workspace layout: z (16*32*16384 f32 = 33.5MB), g (512 f32), wgate (48 f32)
  float* z  = (float*)d_ws;
  float* g  = z + (size_t)B_ * CMID * PIX;
  float* wg = g + (size_t)B_ * CMID;

  proj_kernel       <<<B_ * (PIX / NPIX), 256, 0, stream>>>(x, pw, pb, z);
  mean_kernel       <<<B_ * CMID,         256, 0, stream>>>(z, g);
  gate_kernel       <<<1,                  32, 0, stream>>>(g, fc1w, fc1b, fc2w, fc2b, wg);
  fuse_reproj_kernel<<<B_ * (PIX / NPIX), 256, 0, stream>>>(z, e1w, e1b, e2w, e2b,
                                                            e3w, e3b, wg, rw, rb, out);
}